// TopoGIN_51857435132129
// MI455X (gfx1250) — compile-verified
//
#include <hip/hip_runtime.h>

typedef __attribute__((ext_vector_type(2))) float v2f;
typedef __attribute__((ext_vector_type(8))) float v8f;

#define HD 64
#define BN_EPS 1e-5f

// ---------------------------------------------------------------------------
// Zero a float buffer
// ---------------------------------------------------------------------------
__global__ void zero_kernel(float* __restrict__ p, int n) {
  int i = blockIdx.x * blockDim.x + threadIdx.x;
  if (i < n) p[i] = 0.0f;
}

// ---------------------------------------------------------------------------
// GIN edge aggregation: out[dst[e]] += in[src[e]]  (out pre-seeded with in)
// One thread handles one (edge, 4-feature) chunk: float4 gather + 4 atomics.
// ---------------------------------------------------------------------------
__global__ __launch_bounds__(256) void edge_scatter(
    const float* __restrict__ in, const int* __restrict__ src,
    const int* __restrict__ dst, float* __restrict__ out, int E) {
  long long idx = (long long)blockIdx.x * blockDim.x + threadIdx.x;
  long long total = (long long)E * 16;
  if (idx >= total) return;
  int e = (int)(idx >> 4);
  int c = ((int)idx & 15) * 4;
  int s = src[e];
  int d = dst[e];
  const float4 vv = *(const float4*)(in + (long long)s * HD + c);
  float* o = out + (long long)d * HD + c;
  atomicAdd(o + 0, vv.x);
  atomicAdd(o + 1, vv.y);
  atomicAdd(o + 2, vv.z);
  atomicAdd(o + 3, vv.w);
}

// ---------------------------------------------------------------------------
// global_add_pool: pool[batch[n]] += h[n]
// ---------------------------------------------------------------------------
__global__ __launch_bounds__(256) void pool_scatter(
    const float* __restrict__ in, const int* __restrict__ batch,
    float* __restrict__ pool, int N) {
  long long idx = (long long)blockIdx.x * blockDim.x + threadIdx.x;
  long long total = (long long)N * 16;
  if (idx >= total) return;
  int n = (int)(idx >> 4);
  int c = ((int)idx & 15) * 4;
  int g = batch[n];
  const float4 vv = *(const float4*)(in + (long long)n * HD + c);
  float* o = pool + (long long)g * HD + c;
  atomicAdd(o + 0, vv.x);
  atomicAdd(o + 1, vv.y);
  atomicAdd(o + 2, vv.z);
  atomicAdd(o + 3, vv.w);
}

// ---------------------------------------------------------------------------
// Fused GEMM: out = relu( (A[nrows,64] @ W[64,64]) * scale[c] + shift[c] )
//   mode 0 (BN):  scale = g*rsqrt(v+eps), shift = (bias - m)*scale + be
//   mode 1 (lin): scale = 1,              shift = bias
// One wave per 16-row tile; fp32 WMMA 16x16x4, K=64 in 16 steps, 4 N-tiles.
// A-fragment layout (ISA 7.12.2, 32-bit A 16x4): lanes 0-15 hold K=k0,k0+1;
// lanes 16-31 hold K=k0+2,k0+3 for rows M=lane&15. B mirrors with N=lane&15.
// C/D: VGPR r -> row r (lanes 0-15) / row r+8 (lanes 16-31), col = lane&15.
// ---------------------------------------------------------------------------
__global__ __launch_bounds__(256) void gemm64_fused(
    const float* __restrict__ A, const float* __restrict__ W,
    float* __restrict__ out, int nrows, int mode,
    const float* __restrict__ bias, const float* __restrict__ g,
    const float* __restrict__ be, const float* __restrict__ m,
    const float* __restrict__ v) {
  const int lane = threadIdx.x & 31;
  const int wave = threadIdx.x >> 5;
  const int tile = blockIdx.x * 8 + wave;
  const int ntiles = (nrows + 15) >> 4;
  if (tile >= ntiles) return;  // wave-uniform exit: EXEC stays all-1 for WMMA

  const int row0 = tile << 4;
  const int half = lane >> 4;   // 0: lanes 0-15, 1: lanes 16-31
  const int l = lane & 15;

  int arow = row0 + l;
  if (arow >= nrows) arow = nrows - 1;  // clamped load; extra rows never stored
  const float* Arow = A + (long long)arow * HD;

  // Preload the whole 16x64 A tile into registers (reused across 4 N-tiles).
  v2f afrag[16];
#pragma unroll
  for (int ks = 0; ks < 16; ++ks) {
    const int k = ks * 4 + 2 * half;
    afrag[ks].x = Arow[k];
    afrag[ks].y = Arow[k + 1];
  }

#pragma unroll
  for (int nt = 0; nt < 4; ++nt) {
    const int n0 = nt * 16;
    const int col = n0 + l;
    v8f acc = {};
#pragma unroll
    for (int ks = 0; ks < 16; ++ks) {
      const int k = ks * 4 + 2 * half;
      v2f b;
      b.x = W[k * HD + col];
      b.y = W[(k + 1) * HD + col];
      acc = __builtin_amdgcn_wmma_f32_16x16x4_f32(
          /*neg_a=*/false, afrag[ks], /*neg_b=*/false, b,
          /*c_mod=*/(short)0, acc, /*reuse_a=*/false, /*reuse_b=*/false);
    }

    float scale, shift;
    if (mode == 0) {
      const float s = g[col] * rsqrtf(v[col] + BN_EPS);
      scale = s;
      shift = (bias[col] - m[col]) * s + be[col];
    } else {
      scale = 1.0f;
      shift = bias[col];
    }

#pragma unroll
    for (int r = 0; r < 8; ++r) {
      const int orow = row0 + r + 8 * half;
      float val = acc[r] * scale + shift;
      val = val > 0.0f ? val : 0.0f;
      if (orow < nrows) out[(long long)orow * HD + col] = val;
    }
  }
}

// ---------------------------------------------------------------------------
// Classifier: out[b,c] = bc[c] + pool[b,:] @ Wc[0:64,c] + xt[b,:] @ Wc[64:128,c]
// ---------------------------------------------------------------------------
__global__ __launch_bounds__(256) void classifier(
    const float* __restrict__ pool, const float* __restrict__ xt,
    const float* __restrict__ Wc, const float* __restrict__ bc,
    float* __restrict__ out, int B) {
  int i = blockIdx.x * blockDim.x + threadIdx.x;
  if (i >= B * 10) return;
  int b = i / 10;
  int c = i % 10;
  float acc = bc[c];
  const float* p = pool + (long long)b * HD;
  const float* t = xt + (long long)b * HD;
#pragma unroll 4
  for (int k = 0; k < HD; ++k) acc += p[k] * Wc[k * 10 + c];
#pragma unroll 4
  for (int k = 0; k < HD; ++k) acc += t[k] * Wc[(HD + k) * 10 + c];
  out[i] = acc;
}

// ---------------------------------------------------------------------------
extern "C" void kernel_launch(void* const* d_in, const int* in_sizes, int n_in,
                              void* d_out, int out_size, void* d_ws,
                              size_t ws_size, hipStream_t stream) {
  const float* x    = (const float*)d_in[0];
  const int*   ei   = (const int*)d_in[1];
  const int*   batch= (const int*)d_in[2];
  const float* topo = (const float*)d_in[3];
  const float* W1a  = (const float*)d_in[4];
  const float* b1a  = (const float*)d_in[5];
  const float* g1   = (const float*)d_in[6];
  const float* be1  = (const float*)d_in[7];
  const float* m1   = (const float*)d_in[8];
  const float* v1   = (const float*)d_in[9];
  const float* W1b  = (const float*)d_in[10];
  const float* b1b  = (const float*)d_in[11];
  const float* W2   = (const float*)d_in[12];
  const float* b2   = (const float*)d_in[13];
  const float* g2   = (const float*)d_in[14];
  const float* be2  = (const float*)d_in[15];
  const float* m2   = (const float*)d_in[16];
  const float* v2   = (const float*)d_in[17];
  const float* Wt   = (const float*)d_in[18];
  const float* bt   = (const float*)d_in[19];
  const float* Wc   = (const float*)d_in[20];
  const float* bc   = (const float*)d_in[21];

  const int N = in_sizes[0] / HD;
  const int E = in_sizes[1] / 2;
  const int B = in_sizes[3] / HD;
  const int* src = ei;
  const int* dst = ei + E;

  float* bufA = (float*)d_ws;                 // [N,64]
  float* bufB = bufA + (size_t)N * HD;        // [N,64]
  float* pool = bufB + (size_t)N * HD;        // [B,64]
  float* xt   = pool + (size_t)B * HD;        // [B,64]

  const size_t nbytes = (size_t)N * HD * sizeof(float);
  const long long ethreads = (long long)E * 16;
  const int eblocks = (int)((ethreads + 255) / 256);
  const int gtiles = (N + 15) / 16;
  const int gblocks = (gtiles + 7) / 8;

  // agg1: bufA = x + scatter(x)
  hipMemcpyAsync(bufA, x, nbytes, hipMemcpyDeviceToDevice, stream);
  edge_scatter<<<eblocks, 256, 0, stream>>>(x, src, dst, bufA, E);

  // conv1: Lin -> BN -> ReLU  (bufA -> bufB)
  gemm64_fused<<<gblocks, 256, 0, stream>>>(bufA, W1a, bufB, N, 0, b1a, g1, be1, m1, v1);
  // conv1: Lin -> ReLU (outer relu idempotent)  (bufB -> bufA)
  gemm64_fused<<<gblocks, 256, 0, stream>>>(bufB, W1b, bufA, N, 1, b1b, b1b, b1b, b1b, b1b);

  // agg2: bufB = h3 + scatter(h3)
  hipMemcpyAsync(bufB, bufA, nbytes, hipMemcpyDeviceToDevice, stream);
  edge_scatter<<<eblocks, 256, 0, stream>>>(bufA, src, dst, bufB, E);

  // conv2: Lin -> BN -> ReLU  (bufB -> bufA)
  gemm64_fused<<<gblocks, 256, 0, stream>>>(bufB, W2, bufA, N, 0, b2, g2, be2, m2, v2);

  // global_add_pool
  zero_kernel<<<(B * HD + 255) / 256, 256, 0, stream>>>(pool, B * HD);
  const long long pthreads = (long long)N * 16;
  pool_scatter<<<(int)((pthreads + 255) / 256), 256, 0, stream>>>(bufA, batch, pool, N);

  // topo_net: Lin -> ReLU  (topo -> xt)
  const int ttiles = (B + 15) / 16;
  gemm64_fused<<<(ttiles + 7) / 8, 256, 0, stream>>>(topo, Wt, xt, B, 1, bt, bt, bt, bt, bt);

  // classifier
  classifier<<<(B * 10 + 255) / 256, 256, 0, stream>>>(pool, xt, Wc, bc, (float*)d_out, B);
}